// CLUBLoss_36034775613659
// MI455X (gfx1250) — compile-verified
//
#include <hip/hip_runtime.h>
#include <hip/hip_bf16.h>

// ---------------------------------------------------------------------------
// CLUB loss, fused single-pass, MI455X (gfx1250, wave32).
//   out = -0.5/N * ( P - sum_d( m2*S0 - 2*m1*S1 + S2 ) )
// Traffic: 96 MB (one read of x, p_mu, p_logvar) -> ~4.2 us at 23.3 TB/s.
// x tile is transposed into LDS so BOTH x (token-contiguous) and mu/lv
// (d-contiguous) stream fully coalesced. Staging uses gfx1250 async
// global->LDS ops (ASYNCcnt) when the toolchain exposes them.
// ---------------------------------------------------------------------------

#if defined(__HIP_DEVICE_COMPILE__) && __has_builtin(__builtin_amdgcn_wmma_f32_16x16x4_f32)
#define HAVE_WMMA_F32_16X16X4 1
#endif

// ---- async global->LDS (gfx1250) with guarded fallback --------------------
// Probe-confirmed: builtin exists; pointer params are typed 'int*' in AS1/AS3
// (clang prints them as '__device__ int *' / '__shared__ int *').
#if defined(__HIP_DEVICE_COMPILE__) && __has_builtin(__builtin_amdgcn_global_load_async_to_lds_b32)
#define HAVE_ASYNC_LDS 1
typedef __attribute__((address_space(1))) int as1_int;
typedef __attribute__((address_space(3))) int as3_int;
__device__ __forceinline__ void async_g2l_b32(const float* g, float* l) {
    __builtin_amdgcn_global_load_async_to_lds_b32(
        (as1_int*)(g), (as3_int*)(l), /*offset=*/0, /*cpol=*/0);
}
#if __has_builtin(__builtin_amdgcn_s_wait_asynccnt)
#define ASYNC_WAIT0() __builtin_amdgcn_s_wait_asynccnt(0)
#else
#define ASYNC_WAIT0() asm volatile("s_wait_asynccnt 0" ::: "memory")
#endif
#endif

typedef __attribute__((ext_vector_type(2))) float v2f;
typedef __attribute__((ext_vector_type(8))) float v8f;

// Problem constants (fixed by the reference)
#define BB        16
#define DD        512
#define HWW       1024                  // H*W
#define DHW       (DD * HWW)            // 524288
#define NTOK      (BB * HWW)            // 16384
#define TILE_T    64                    // tokens per tile (all in same b: 64 | 1024)
#define NTILES    (NTOK / TILE_T)       // 256
#define DCHUNK    128
#define NCHUNK    (DD / DCHUNK)         // 4
#define XT_STRIDE (DCHUNK + 4)          // 132: rows 16B-aligned, banks spread
#define PER_BLK   (5 * DD + 1)          // ws floats per block slice (2561)

// ---------------------------------------------------------------------------
__global__ void club_zero(float* __restrict__ ws, int n) {
    int i = blockIdx.x * blockDim.x + threadIdx.x;
    if (i < n) ws[i] = 0.0f;
}

// ---------------------------------------------------------------------------
// Main streaming kernel: 256 threads (8 wave32), grid-stride over tiles.
// Per chunk: stage x-tile transposed to LDS, then lanes sweep d (mu/lv
// coalesced), accumulating P (scalar) and per-d {T1,T2,S0,S1,S2}.
// ---------------------------------------------------------------------------
__global__ __launch_bounds__(256)
void club_main(const float* __restrict__ x,
               const float* __restrict__ mu,
               const float* __restrict__ lv,
               float* __restrict__ ws, int nblk) {
    __shared__ float xt[TILE_T * XT_STRIDE];   // 64*132*4 = 33,792 B
    __shared__ float red[5 * 8 * DCHUNK];      // 5*8*128*4 = 20,480 B

    const int tid  = threadIdx.x;
    const int wave = tid >> 5;
    const int lane = tid & 31;

    float* wsT1 = ws;
    float* wsT2 = ws + (size_t)nblk * DD;
    float* wsS0 = ws + (size_t)nblk * DD * 2;
    float* wsS1 = ws + (size_t)nblk * DD * 3;
    float* wsS2 = ws + (size_t)nblk * DD * 4;
    float* wsP  = ws + (size_t)nblk * DD * 5;

    float accP = 0.0f;

    for (int tile = blockIdx.x; tile < NTILES; tile += nblk) {
        const int tok0 = tile * TILE_T;
        const int b    = tok0 >> 10;        // / 1024
        const int hw0  = tok0 & 1023;

        for (int c = 0; c < NCHUNK; ++c) {
            const int d0 = c * DCHUNK;

            // ---- stage: x[64 tok][128 d] -> LDS transposed -------------
#ifdef HAVE_ASYNC_LDS
            {
                // Per-lane async scatter: global contiguous along t (lanes =
                // 32 consecutive floats -> 128B requests), LDS contiguous
                // along d. Hardware does the transpose; no VGPR round-trip.
                const int t  = tid & 63;        // token
                const int dg = tid >> 6;        // 0..3 d-row group
                const float* gbase = x + (size_t)b * DHW + hw0 + t;
                float*       lbase = xt + t * XT_STRIDE;
                #pragma unroll
                for (int p = 0; p < 32; ++p) {
                    const int dl = p * 4 + dg;  // 0..127
                    async_g2l_b32(gbase + (size_t)(d0 + dl) * HWW, lbase + dl);
                }
                ASYNC_WAIT0();
            }
#else
            {
                const int t4 = (tid & 15) * 4;   // 4-token group
                const int dd = tid >> 4;         // 16 d-rows per pass
                #pragma unroll
                for (int p = 0; p < 8; ++p) {
                    const int dl = p * 16 + dd;  // 0..127
                    const float4 v = *(const float4*)(
                        x + (size_t)b * DHW + (size_t)(d0 + dl) * HWW + hw0 + t4);
                    xt[(t4 + 0) * XT_STRIDE + dl] = v.x;
                    xt[(t4 + 1) * XT_STRIDE + dl] = v.y;
                    xt[(t4 + 2) * XT_STRIDE + dl] = v.z;
                    xt[(t4 + 3) * XT_STRIDE + dl] = v.w;
                }
            }
#endif
            __syncthreads();

            // ---- compute: wave handles 8 tokens, lanes sweep d ----------
            float4 t1v = {0,0,0,0}, t2v = {0,0,0,0};
            float4 s0v = {0,0,0,0}, s1v = {0,0,0,0}, s2v = {0,0,0,0};
            #pragma unroll
            for (int j = 0; j < 8; ++j) {
                const int t = wave * 8 + j;
                const size_t i = (size_t)(tok0 + t);
                const float4 xv  = *(const float4*)(xt + t * XT_STRIDE + lane * 4);
                const float4 muv = *(const float4*)(mu + i * DD + d0 + lane * 4);
                const float4 lvv = *(const float4*)(lv + i * DD + d0 + lane * 4);
                #define ACC(C) { float dx_ = xv.C - muv.C;                    \
                                 float iv_ = __expf(-lvv.C);                  \
                                 accP  += dx_ * dx_ * iv_;                    \
                                 s0v.C += iv_;                                \
                                 s1v.C += muv.C * iv_;                        \
                                 s2v.C += muv.C * muv.C * iv_;                \
                                 t1v.C += xv.C;                               \
                                 t2v.C += xv.C * xv.C; }
                ACC(x) ACC(y) ACC(z) ACC(w)
                #undef ACC
            }

            // ---- cross-wave reduce of per-d accumulators ---------------
            *(float4*)(red + 0 * 1024 + wave * 128 + lane * 4) = t1v;
            *(float4*)(red + 1 * 1024 + wave * 128 + lane * 4) = t2v;
            *(float4*)(red + 2 * 1024 + wave * 128 + lane * 4) = s0v;
            *(float4*)(red + 3 * 1024 + wave * 128 + lane * 4) = s1v;
            *(float4*)(red + 4 * 1024 + wave * 128 + lane * 4) = s2v;
            __syncthreads();

            if (tid < DCHUNK) {
                float a0 = 0, a1 = 0, a2 = 0, a3 = 0, a4 = 0;
                #pragma unroll
                for (int w2 = 0; w2 < 8; ++w2) {
                    a0 += red[0 * 1024 + w2 * 128 + tid];
                    a1 += red[1 * 1024 + w2 * 128 + tid];
                    a2 += red[2 * 1024 + w2 * 128 + tid];
                    a3 += red[3 * 1024 + w2 * 128 + tid];
                    a4 += red[4 * 1024 + w2 * 128 + tid];
                }
                const size_t dg = (size_t)blockIdx.x * DD + d0 + tid;
                wsT1[dg] += a0;  wsT2[dg] += a1;
                wsS0[dg] += a2;  wsS1[dg] += a3;  wsS2[dg] += a4;
            }
            __syncthreads();   // protects xt (next stage) and red (next write)
        }
    }

    // ---- block-reduce P ----------------------------------------------------
    red[tid] = accP;
    __syncthreads();
    #pragma unroll
    for (int s = 128; s > 0; s >>= 1) {
        if (tid < s) red[tid] += red[tid + s];
        __syncthreads();
    }
    if (tid == 0) wsP[blockIdx.x] = red[0];
}

// ---------------------------------------------------------------------------
// Finalize: 1 block x 512 threads (thread == d). Fold per-block partials,
// form per-d combined value, reduce 512 floats with v_wmma_f32_16x16x4_f32
// (B = ones -> row sums), write scalar.
// ---------------------------------------------------------------------------
__global__ __launch_bounds__(512)
void club_final(const float* __restrict__ ws, float* __restrict__ out, int nblk) {
    __shared__ float val[DD];
    const int tid = threadIdx.x;           // == d

    const float* wsT1 = ws;
    const float* wsT2 = ws + (size_t)nblk * DD;
    const float* wsS0 = ws + (size_t)nblk * DD * 2;
    const float* wsS1 = ws + (size_t)nblk * DD * 3;
    const float* wsS2 = ws + (size_t)nblk * DD * 4;
    const float* wsP  = ws + (size_t)nblk * DD * 5;

    float sT1 = 0, sT2 = 0, sS0 = 0, sS1 = 0, sS2 = 0;
    for (int blk = 0; blk < nblk; ++blk) {
        const size_t o = (size_t)blk * DD + tid;
        sT1 += wsT1[o];  sT2 += wsT2[o];
        sS0 += wsS0[o];  sS1 += wsS1[o];  sS2 += wsS2[o];
    }
    const float invN = 1.0f / (float)NTOK;
    const float m1 = sT1 * invN;
    const float m2 = sT2 * invN;
    const float negd = m2 * sS0 - 2.0f * m1 * sS1 + sS2;

    float p = 0.0f;
    for (int k = tid; k < nblk; k += 512) p += wsP[k];

    val[tid] = p - negd;                   // sum over tid == P - NEG
    __syncthreads();

    if (tid < 32) {                        // wave 0, EXEC all-ones
        float tot;
#ifdef HAVE_WMMA_F32_16X16X4
        v8f c = {};
        v2f ones; ones[0] = 1.0f; ones[1] = 1.0f;
        #pragma unroll
        for (int q = 0; q < 8; ++q) {
            // 32-bit A 16x4 layout: lanes 0-15 -> K=0,1; lanes 16-31 -> K=2,3
            const int idx = q * 64 + 4 * (tid & 15) + 2 * (tid >> 4);
            v2f a; a[0] = val[idx]; a[1] = val[idx + 1];
            c = __builtin_amdgcn_wmma_f32_16x16x4_f32(
                false, a, false, ones, (short)0, c, false, false);
        }
        float s = c[0] + c[1] + c[2] + c[3] + c[4] + c[5] + c[6] + c[7];
        tot = s + __shfl_xor(s, 16, 32);   // M=0..7 half + M=8..15 half
#else
        float s = 0.0f;
        for (int k = tid; k < DD; k += 32) s += val[k];
        #pragma unroll
        for (int off = 16; off > 0; off >>= 1) s += __shfl_xor(s, off, 32);
        tot = s;
#endif
        if (tid == 0) out[0] = -0.5f * tot * invN;
    }
}

// ---------------------------------------------------------------------------
extern "C" void kernel_launch(void* const* d_in, const int* in_sizes, int n_in,
                              void* d_out, int out_size, void* d_ws, size_t ws_size,
                              hipStream_t stream) {
    (void)in_sizes; (void)n_in; (void)out_size;
    const float* x  = (const float*)d_in[0];
    const float* mu = (const float*)d_in[1];
    const float* lv = (const float*)d_in[2];
    float* out = (float*)d_out;
    float* ws  = (float*)d_ws;

    int nblk = NTILES;                               // 256 preferred
    const size_t cap = ws_size / sizeof(float);
    if ((size_t)nblk * PER_BLK > cap) {              // degrade if ws is small
        nblk = (int)(cap / PER_BLK);
        if (nblk < 1) nblk = 1;
        if (nblk > NTILES) nblk = NTILES;
    }

    const int zn = nblk * PER_BLK;
    club_zero <<<(zn + 255) / 256, 256, 0, stream>>>(ws, zn);
    club_main <<<nblk, 256, 0, stream>>>(x, mu, lv, ws, nblk);
    club_final<<<1, 512, 0, stream>>>(ws, out, nblk);
}